// InstanceMap_60876866453670
// MI455X (gfx1250) — compile-verified
//
#include <hip/hip_runtime.h>
#include <stdint.h>

// Problem constants (from reference): B, N_INST, C, K, H, W = 4, 64, 16, 4, 480, 480
#define BB     4
#define NINST  64
#define CC     16
#define KK     4
#define HH     480
#define WW     480
#define HWSZ   (HH * WW)        // 230400, divisible by TILE
#define TILE   256
#define BIGID  1073741824       // 2^30, matches reference BIG

// Tiny scratch layout (ints) at the start of d_ws; agg byte-labels at +1024B.
#define WS_CHMAX 0                      // [B*C]  running per-channel max of gmap
#define WS_OVER  (WS_CHMAX + BB * CC)   // [B*K]  per-step segment min (overlap)
#define WS_FLAGS (WS_OVER + BB * KK)    // [B]    bits 0..3 = present_k, bit 4 = any
#define WS_MAP   (WS_FLAGS + BB)        // [B*K]  per-step mapping values
#define WS_UPD   (WS_MAP + BB * KK)     // [B]    per-step "channel updated" flag
#define WS_TOTAL (WS_UPD + BB)          // 104 ints

// ---- CDNA5 Tensor Data Mover availability (device pass only) ----
#if defined(__has_builtin)
#if __has_builtin(__builtin_amdgcn_tensor_load_to_lds) && \
    __has_builtin(__builtin_amdgcn_s_wait_tensorcnt)
#define USE_TDM 1
#endif
#endif

typedef uint32_t u32x4 __attribute__((ext_vector_type(4)));
typedef int      i32x4 __attribute__((ext_vector_type(4)));
typedef int      i32x8 __attribute__((ext_vector_type(8)));

// Stage a 2D tile (TILE pixels x KK channels, channel stride HWSZ floats) from
// global memory into LDS:  ltile[k*TILE + x] = gsrc[k*HWSZ + x].
// Uses the TDM (tensor_load_to_lds) when available; ends with a block barrier.
__device__ __forceinline__ void stage_tile4(const float* gsrc, float* ltile) {
#ifdef USE_TDM
  if (threadIdx.x < 32) {  // one TDM issue per block (wave 0)
    uint64_t ga = (uint64_t)(uintptr_t)gsrc;        // global byte address
    uint32_t la = (uint32_t)(uintptr_t)ltile;       // low 32 bits = LDS offset
    // D# group 0: count=1 | lds_addr | global_addr[56:0] | type=2
    u32x4 g0 = { 1u,
                 la,
                 (uint32_t)ga,
                 (uint32_t)((ga >> 32) & 0x01FFFFFFu) | (2u << 30) };
    // D# group 1: data_size=4B; tensor_dim0=HWSZ; tensor_dim1=KK;
    //             tile_dim0=TILE; tile_dim1=KK; tensor_dim0_stride=HWSZ
    i32x8 g1 = { (int)0x00020000u,                                  // data_size=2 (4B)
                 (int)(((unsigned)HWSZ & 0xFFFFu) << 16),           // dim0[15:0]<<16
                 (int)(((unsigned)HWSZ >> 16) | ((unsigned)KK << 16)), // dim0[31:16]|dim1lo<<16
                 (int)((unsigned)TILE << 16),                       // dim1hi=0 | tile_dim0<<16
                 (int)KK,                                           // tile_dim1 | tile_dim2=0
                 (int)HWSZ,                                         // dim0_stride[31:0]
                 0, 0 };
    i32x4 z4 = { 0, 0, 0, 0 };
#if __clang_major__ >= 23
    i32x8 z8 = { 0, 0, 0, 0, 0, 0, 0, 0 };
    __builtin_amdgcn_tensor_load_to_lds(g0, g1, z4, z4, z8, 0);
#else
    __builtin_amdgcn_tensor_load_to_lds(g0, g1, z4, z4, 0);
#endif
    __builtin_amdgcn_s_wait_tensorcnt(0);
  }
#else
  for (int i = threadIdx.x; i < KK * TILE; i += TILE) {
    int k = i / TILE, x = i % TILE;
    ltile[i] = gsrc[(size_t)k * HWSZ + x];
  }
#endif
  __syncthreads();
}

// argmax over the 4 staged channels with reference tie-breaking (first max),
// background wins unless max > 1e-5. Returns the instance label (0 = bg).
__device__ __forceinline__ int compute_label(const float* lt, int t,
                                             const int* idrow) {
  float m = lt[t];
  int kb = 0;
#pragma unroll
  for (int k = 1; k < KK; ++k) {
    float u = lt[k * TILE + t];
    if (u > m) { m = u; kb = k; }
  }
  return (m > 1e-5f) ? idrow[kb] : 0;
}

// ---------------- kernels ----------------

__global__ void k_ws_init(int* ws) {
  int t = blockIdx.x * blockDim.x + threadIdx.x;
  if (t < WS_TOTAL) ws[t] = 0;
}

// out = init map; build per-channel running max table.
__global__ void k_copy_chmax(const int* __restrict__ initmap,
                             int* __restrict__ out, int* __restrict__ ws) {
  int t = threadIdx.x, c = blockIdx.y, b = blockIdx.z;
  size_t off = ((size_t)b * CC + c) * HWSZ + blockIdx.x * TILE + t;
  int v = initmap[off];
  out[off] = v;
  __shared__ int smax;
  if (t == 0) smax = 0;
  __syncthreads();
  atomicMax(&smax, v);           // LDS atomic (ds_max)
  __syncthreads();
  if (t == 0) atomicMax(&ws[WS_CHMAX + b * CC + c], smax);
}

// Precompute 1-byte labels agg[b,c,p] (read 236MB of update_map exactly once).
__global__ void k_agg(const float* __restrict__ um, const int* __restrict__ ids,
                      unsigned char* __restrict__ aggp) {
  int t = threadIdx.x, c = blockIdx.y, b = blockIdx.z;
  int p0 = blockIdx.x * TILE;
  __shared__ float ltile[KK * TILE];
  const int* idrow = ids + ((size_t)b * CC + c) * KK;
  const float* gsrc = um + ((size_t)b * NINST + idrow[0]) * HWSZ + p0;
  stage_tile4(gsrc, ltile);
  int label = compute_label(ltile, t, idrow);
  aggp[((size_t)b * CC + c) * HWSZ + p0 + t] = (unsigned char)label;
}

__global__ void k_step_reset(int* ws) {
  int t = threadIdx.x;
  if (t < BB * KK) ws[WS_OVER + t] = BIGID;
  if (t < BB) ws[WS_FLAGS + t] = 0;
}

// Per-step segment reductions: overlap_min per k, present bits, any-flag.
__global__ void k_reduce(const float* __restrict__ um, const int* __restrict__ ids,
                         const int* __restrict__ out, int* __restrict__ ws,
                         const unsigned char* __restrict__ aggp, int ci, int useAgg) {
  int t = threadIdx.x, b = blockIdx.z;
  int p0 = blockIdx.x * TILE;
  __shared__ float ltile[KK * TILE];
  __shared__ int s_min[KK];
  __shared__ int s_flags;
  if (t < KK) s_min[t] = BIGID;
  if (t == 0) s_flags = 0;
  const int* idrow = ids + ((size_t)b * CC + ci) * KK;
  int n0 = idrow[0];
  int label;
  if (useAgg) {
    __syncthreads();
    label = aggp[((size_t)b * CC + ci) * HWSZ + p0 + t];
  } else {
    const float* gsrc = um + ((size_t)b * NINST + n0) * HWSZ + p0;
    stage_tile4(gsrc, ltile);
    label = compute_label(ltile, t, idrow);
  }
  if (label != 0) {
    int k = label - n0;                       // ids are ascending per (c,k)
    int g = out[((size_t)b * CC + ci) * HWSZ + p0 + t];
    atomicMin(&s_min[k], (g != 0) ? g : BIGID);
    atomicOr(&s_flags, (1 << k) | 16);
  }
  __syncthreads();
  if (t < KK && s_min[t] < BIGID) atomicMin(&ws[WS_OVER + b * KK + t], s_min[t]);
  if (t == 0 && s_flags) atomicOr(&ws[WS_FLAGS + b], s_flags);
}

// Tiny per-batch mapping: new-id assignment vs running global max; also
// updates the running channel max (replaces jnp.max(gmap) each step).
__global__ void k_mapping(const int* __restrict__ ids, int* __restrict__ ws, int ci) {
  int b = threadIdx.x;
  if (b >= BB) return;
  int maxid = 0;
  for (int c = 0; c < CC; ++c) maxid = max(maxid, ws[WS_CHMAX + b * CC + c]);
  const int* idrow = ids + ((size_t)b * CC + ci) * KK;
  int flags = ws[WS_FLAGS + b];
  int run = 0, chNew = 0;
  for (int k = 0; k < KK; ++k) {
    int l = idrow[k];
    int mp = 0;
    if (l != 0 && ((flags >> k) & 1)) {       // present & not the null label
      int ov = ws[WS_OVER + b * KK + k];
      if (ov < BIGID) mp = ov;                // overlap -> take min existing id
      else { run += 1; mp = maxid + run; }    // brand-new instance id
    }
    ws[WS_MAP + b * KK + k] = mp;
    if ((flags >> k) & 1) chNew = max(chNew, mp);
  }
  int upd = (flags & 16) ? 1 : 0;
  ws[WS_UPD + b] = upd;
  if (upd) {
    int cur = ws[WS_CHMAX + b * CC + ci];
    ws[WS_CHMAX + b * CC + ci] = max(cur, chNew);
  }
}

// Apply glob = max(glob, mapping[local]) on channel ci (if anything present).
__global__ void k_update(const float* __restrict__ um, const int* __restrict__ ids,
                         int* __restrict__ out, const int* __restrict__ ws,
                         const unsigned char* __restrict__ aggp, int ci, int useAgg) {
  int t = threadIdx.x, b = blockIdx.z;
  if (ws[WS_UPD + b] == 0) return;            // uniform per block
  int p0 = blockIdx.x * TILE;
  __shared__ float ltile[KK * TILE];
  const int* idrow = ids + ((size_t)b * CC + ci) * KK;
  int n0 = idrow[0];
  int label;
  if (useAgg) {
    label = aggp[((size_t)b * CC + ci) * HWSZ + p0 + t];
  } else {
    const float* gsrc = um + ((size_t)b * NINST + n0) * HWSZ + p0;
    stage_tile4(gsrc, ltile);
    label = compute_label(ltile, t, idrow);
  }
  if (label != 0) {
    int mp = ws[WS_MAP + b * KK + (label - n0)];
    size_t off = ((size_t)b * CC + ci) * HWSZ + p0 + t;
    int g = out[off];
    if (mp > g) out[off] = mp;
  }
}

extern "C" void kernel_launch(void* const* d_in, const int* in_sizes, int n_in,
                              void* d_out, int out_size, void* d_ws, size_t ws_size,
                              hipStream_t stream) {
  const float* um      = (const float*)d_in[0];  // (B, N_INST, H, W) f32
  const int*   initmap = (const int*)d_in[1];    // (B, C, H, W) i32
  const int*   ids     = (const int*)d_in[2];    // (B, C, K) i32
  int* out = (int*)d_out;                        // (B, C, H, W) i32
  int* ws  = (int*)d_ws;

  // Use byte-label agg cache if workspace is large enough; else recompute.
  size_t aggBytes = (size_t)BB * CC * HWSZ;
  int useAgg = (ws_size >= 1024 + aggBytes) ? 1 : 0;
  unsigned char* aggp = (unsigned char*)d_ws + 1024;

  k_ws_init<<<1, 128, 0, stream>>>(ws);

  dim3 gAll(HWSZ / TILE, CC, BB);
  k_copy_chmax<<<gAll, TILE, 0, stream>>>(initmap, out, ws);
  if (useAgg) k_agg<<<gAll, TILE, 0, stream>>>(um, ids, aggp);

  dim3 gCh(HWSZ / TILE, 1, BB);
  for (int ci = 0; ci < CC; ++ci) {
    k_step_reset<<<1, 32, 0, stream>>>(ws);
    k_reduce<<<gCh, TILE, 0, stream>>>(um, ids, out, ws, aggp, ci, useAgg);
    k_mapping<<<1, 32, 0, stream>>>(ids, ws, ci);
    k_update<<<gCh, TILE, 0, stream>>>(um, ids, out, ws, aggp, ci, useAgg);
  }
}